// StreamingDurationProjector_51110110822753
// MI455X (gfx1250) — compile-verified
//
#include <hip/hip_runtime.h>

// StreamingDurationProjector: per-row nonlinear scan over N steps.
// One thread per row; 32-row x 32-col tiles of the 5 [B,N] inputs are staged
// into LDS (double-buffered) via CDNA5 async global->LDS DMA so global reads
// are coalesced even though each lane walks one row.

#define TILE   32
#define STRIDE 33      // +1 pad -> odd stride: conflict-free lane-varying-row reads
#define NARR   5

#if __has_builtin(__builtin_amdgcn_global_load_async_to_lds_b32)
#define HAVE_ASYNC 1
#else
#define HAVE_ASYNC 0
#endif

typedef __attribute__((address_space(1))) int gas_int;   // global (device) word
typedef __attribute__((address_space(3))) int las_int;   // LDS word

__global__ __launch_bounds__(32)
void duration_projector_kernel(const float* __restrict__ ex_g,
                               const float* __restrict__ src_g,
                               const float* __restrict__ act_g,
                               const float* __restrict__ sp_g,
                               const float* __restrict__ bc_g,
                               const float* __restrict__ carry0_g,
                               const float* __restrict__ poff0_g,
                               float* __restrict__ out,
                               int B, int N)
{
    __shared__ float sbuf[2 * NARR * TILE * STRIDE];

    const int lane    = threadIdx.x;            // 0..31, one row per lane
    const int rowBase = blockIdx.x * TILE;
    const int row     = rowBase + lane;

    const float* gin[NARR] = {ex_g, src_g, act_g, sp_g, bc_g};

    const size_t BN = (size_t)B * (size_t)N;
    float* proj_o  = out;
    float* carry_o = out + BN;
    float* poff_o  = out + BN + (size_t)B;
    float* hit_o   = out + BN + 2 * (size_t)B;
    float* decay_o = out + 2 * BN + 2 * (size_t)B;

    float carry = carry0_g[row];
    float poff  = poff0_g[row];

    // ---- tile staging: each async issue copies one coalesced 128B row chunk
    auto issue_tile = [&](int buf, int t0) {
        float* base = &sbuf[buf * NARR * TILE * STRIDE];
#if HAVE_ASYNC
        #pragma unroll
        for (int a = 0; a < NARR; ++a) {
            const float* g = gin[a] + (size_t)rowBase * N + t0 + lane;
            float*       l = base + a * TILE * STRIDE + lane;
            #pragma unroll
            for (int r = 0; r < TILE; ++r) {
                __builtin_amdgcn_global_load_async_to_lds_b32(
                    (gas_int*)(g + (size_t)r * N),
                    (las_int*)(l + r * STRIDE),
                    /*offset=*/0, /*cpol=*/0);
            }
        }
#else
        #pragma unroll
        for (int a = 0; a < NARR; ++a) {
            const float* g = gin[a] + (size_t)rowBase * N + t0 + lane;
            float*       l = base + a * TILE * STRIDE + lane;
            #pragma unroll 4
            for (int r = 0; r < TILE; ++r) {
                l[r * STRIDE] = g[(size_t)r * N];
                if (t0 + TILE < N)
                    __builtin_prefetch(g + (size_t)r * N + TILE, 0, 1);
            }
        }
#endif
    };

    auto wait_tile = [&]() {
#if HAVE_ASYNC
#if __has_builtin(__builtin_amdgcn_s_wait_asynccnt)
        __builtin_amdgcn_s_wait_asynccnt(0);
#else
        asm volatile("s_wait_asynccnt 0" ::: "memory");
#endif
        asm volatile("" ::: "memory");   // keep ds reads after the wait
#else
        __syncthreads();                 // single-wave WG: cheap
#endif
    };

    issue_tile(0, 0);
    int buf = 0;

    for (int t0 = 0; t0 < N; t0 += TILE) {
        wait_tile();
        if (t0 + TILE < N) issue_tile(buf ^ 1, t0 + TILE);

        const float* base = &sbuf[buf * NARR * TILE * STRIDE + lane * STRIDE];

        for (int t = 0; t < TILE; ++t) {
            const float vex  = base[0 * TILE * STRIDE + t];
            const float vsrc = base[1 * TILE * STRIDE + t];
            const float vact = base[2 * TILE * STRIDE + t];
            const float vsp  = base[3 * TILE * STRIDE + t];
            const float vbc  = base[4 * TILE * STRIDE + t];

            const bool active = vact > 0.5f;
            const bool speech = vsp  > 0.5f;

            const float src_cnt = fmaxf(0.0f, floorf(vsrc + 0.5f));
            const float total   = fmaxf(0.0f, vex + carry);
            float frames        = fmaxf(1.0f, floorf(total + 0.5f));
            const float anchor  = fmaxf(1.0f, src_cnt);
            const float lower   = fmaxf(1.0f,  ceilf(anchor - (24.0f + poff)));
            const float upper   = fmaxf(lower, floorf(anchor + (24.0f - poff)));
            frames = fminf(fmaxf(frames, lower), upper);

            float poff_s  = poff + frames - anchor;
            float carry_s = total - frames;

            const bool boundary = vbc >= 0.5f;
            if (boundary) {
                carry_s *= 0.25f;
                poff_s = fminf(fmaxf(poff_s * 0.25f, -24.0f), 24.0f);
            }

            const float hitv  = (boundary && speech && active) ? 1.0f : 0.0f;
            const float projv = active ? (speech ? frames : src_cnt) : 0.0f;

            if (speech && active) { carry = carry_s; poff = poff_s; }

            const size_t o = (size_t)row * N + (size_t)(t0 + t);
            proj_o[o]  = projv;
            hit_o[o]   = hitv;
            decay_o[o] = hitv;   // APPLY_DECAY is statically true -> decay == hit
        }
        buf ^= 1;
    }

    carry_o[row] = carry;
    poff_o[row]  = poff;
}

extern "C" void kernel_launch(void* const* d_in, const int* in_sizes, int n_in,
                              void* d_out, int out_size, void* d_ws, size_t ws_size,
                              hipStream_t stream) {
    (void)n_in; (void)out_size; (void)d_ws; (void)ws_size;
    const float* ex   = (const float*)d_in[0];
    const float* src  = (const float*)d_in[1];
    const float* act  = (const float*)d_in[2];
    const float* sp   = (const float*)d_in[3];
    const float* bc   = (const float*)d_in[4];
    const float* res0 = (const float*)d_in[5];
    const float* pof0 = (const float*)d_in[6];

    const int B = in_sizes[5];            // residual_prev is [B,1]
    const int N = in_sizes[0] / B;        // [B,N] inputs

    dim3 grid(B / TILE), block(TILE);
    duration_projector_kernel<<<grid, block, 0, stream>>>(
        ex, src, act, sp, bc, res0, pof0, (float*)d_out, B, N);
}